// QuantumGate_6485400617324
// MI455X (gfx1250) — compile-verified
//
#include <hip/hip_runtime.h>
#include <math.h>

// M = cos(a)*I - i*sin(a)*P, P[i,j]=1 iff j == i ^ MASK, N = 4096.
// Output is complex64 viewed as interleaved float32 (re, im): 128 MB.
// Pure store-bandwidth problem; output fits in the 192MB L2 so default
// (RT) temporal stores ack at L2. Bulk zeros go out via the CDNA5 async
// LDS->global store path (ASYNCcnt), then the two nonzero complex slots
// per row are patched after s_wait_asynccnt + barrier.

#define NQ_DIM   4096
#define XX_MASK  272            // bits 8 and 4 (qubits 3 and 7, MSB-first kron)
#define ROW_F4   2048           // float4 chunks per row: 4096 complex * 8B / 16B

typedef float f4  __attribute__((ext_vector_type(4)));
typedef int   v4i __attribute__((ext_vector_type(4)));

// Builtin expects: (v4i addrspace(1)*, v4i addrspace(3)*, int offset, int cpol)
typedef __attribute__((address_space(1))) v4i v4i_glb;
typedef __attribute__((address_space(3))) v4i v4i_lds;

#ifndef __has_builtin
#define __has_builtin(x) 0
#endif

#if __has_builtin(__builtin_amdgcn_global_store_async_from_lds_b128)
#define USE_ASYNC_LDS 1
#else
#define USE_ASYNC_LDS 0
#endif

__global__ __launch_bounds__(256) void qgate_xx_expm_kernel(
    const float* __restrict__ angle, float* __restrict__ out) {
  const int row = blockIdx.x;           // one block per matrix row
  const int t   = threadIdx.x;

  f4* __restrict__ rowp = reinterpret_cast<f4*>(out) + (size_t)row * ROW_F4;

  const int dcol   = row;               // diagonal column -> (cos a, 0)
  const int ocol   = row ^ XX_MASK;     // permutation column -> (0, -sin a)
  const int dchunk = dcol >> 1;         // float4 chunk holding diag complex
  const int ochunk = ocol >> 1;         // chunk holding off-diag complex (!= dchunk)

#if USE_ASYNC_LDS
  // --- bulk zero fill: async LDS->global b128 stores (ASYNCcnt path) ---
  __shared__ v4i zpool[64];             // 1KB zero pool, 16B per lane slot
  if (t < 64) zpool[t] = (v4i){0, 0, 0, 0};
  __syncthreads();

  v4i_lds* zsrc = (v4i_lds*)&zpool[t & 63];
#pragma unroll
  for (int k = 0; k < 8; ++k) {
    const int chunk = t + (k << 8);     // full-EXEC, uniform issue per wave
    __builtin_amdgcn_global_store_async_from_lds_b128(
        (v4i_glb*)(rowp + chunk), zsrc, /*offset=*/0, /*cpol=*/0);
  }
  // Drain this wave's async stores, then barrier so every wave's zeros for
  // this row have landed before the patch overwrites (async vs regular
  // stores are otherwise unordered).
  asm volatile("s_wait_asynccnt 0x0" ::: "memory");
  __syncthreads();

  if (t == 0) {
    const float a = *angle;
    float2* rc = reinterpret_cast<float2*>(rowp);   // complex slots
    rc[dcol] = make_float2(cosf(a), 0.f);
    rc[ocol] = make_float2(0.f, -sinf(a));
  }
#else
  // --- fallback: per-thread patched 128-bit stores (default RT temporal,
  // dirty lines stay in the 192MB L2) ---
  const float a = *angle;
  const float c = cosf(a);
  const float s = sinf(a);
#pragma unroll
  for (int k = 0; k < 8; ++k) {
    const int chunk = t + (k << 8);
    f4 v = {0.f, 0.f, 0.f, 0.f};
    if (chunk == dchunk) v[(dcol & 1) * 2]     = c;    // real part
    if (chunk == ochunk) v[(ocol & 1) * 2 + 1] = -s;   // imag part
    rowp[chunk] = v;                    // global_store_b128
  }
#endif
}

extern "C" void kernel_launch(void* const* d_in, const int* in_sizes, int n_in,
                              void* d_out, int out_size, void* d_ws, size_t ws_size,
                              hipStream_t stream) {
  (void)in_sizes; (void)n_in; (void)d_ws; (void)ws_size; (void)out_size;
  const float* angle = (const float*)d_in[0];   // d_in[1] is the static size (12)
  float* out = (float*)d_out;                   // complex64 as interleaved float32
  qgate_xx_expm_kernel<<<dim3(NQ_DIM), dim3(256), 0, stream>>>(angle, out);
}